// GAT_4088808866112
// MI455X (gfx1250) — compile-verified
//
#include <hip/hip_runtime.h>

// ---------------------------------------------------------------------------
// Two-layer GAT (PyG GATConv semantics) for MI455X / gfx1250.
//   L1: H1 = x@W1 (fp32 WMMA 16x16x4), per-head additive attention, segment
//       softmax over dst (max-shift dropped: softmax is shift invariant),
//       float4 scatter-add aggregation, +b1, ReLU.
//   L2: H2 = h@W2 (fp32 WMMA, weights pre-padded 40->48 cols), heads=1
//       attention, softmax, scatter-add into d_out (pre-initialized with b2).
// Weights are pre-transposed (and zero-padded) so the WMMA inner loop is pure
// unguarded global_load_b64 + v_wmma, fully unrolled over K=128.
// ---------------------------------------------------------------------------

typedef __attribute__((ext_vector_type(2))) float v2f;
typedef __attribute__((ext_vector_type(8))) float v8f;

#define F_IN  128
#define KDIM  128          // inner dimension of both GEMMs
#define HID   32
#define HEADS 4
#define CLS   40
#define CLSP  48           // CLS padded to col-tile multiple
#define NEG_SLOPE 0.2f

static __device__ __forceinline__ void atomAddF32(float* p, float v) {
  // relaxed agent-scope fadd -> non-returning global_atomic_add_f32
  __hip_atomic_fetch_add(p, v, __ATOMIC_RELAXED, __HIP_MEMORY_SCOPE_AGENT);
}

// ---------------------------------------------------------------------------
// WT[c, k] = (c < Ncol) ? W[k, c] : 0       (WT is [NcolPad x K], zero padded)
// ---------------------------------------------------------------------------
__global__ void transpose_pad(const float* __restrict__ W, float* __restrict__ WT,
                              int K, int Ncol, int NcolPad)
{
  const int idx = blockIdx.x * blockDim.x + threadIdx.x;
  const int tot = K * NcolPad;
  if (idx >= tot) return;
  const int c = idx / K;
  const int k = idx % K;
  WT[(size_t)c * K + k] = (c < Ncol) ? W[(size_t)k * Ncol + c] : 0.0f;
}

// ---------------------------------------------------------------------------
// C[M,Ncol] = A[M,128] @ BT^T, BT is [NcolPad x 128] (pre-transposed, padded).
// One wave per 16x16 output tile; K stepped by 4 with V_WMMA_F32_16X16X4_F32.
// Requires M % 16 == 0. No predication in the inner loop.
// ---------------------------------------------------------------------------
__global__ __launch_bounds__(256)
void wmma_gemm_f32(const float* __restrict__ A, const float* __restrict__ BT,
                   float* __restrict__ C, int M, int Ncol, int nColTiles)
{
  const int waveId = blockIdx.x * (blockDim.x >> 5) + (threadIdx.x >> 5);
  const int lane   = threadIdx.x & 31;
  const int totalTiles = (M >> 4) * nColTiles;
  if (waveId >= totalTiles) return;            // wave-uniform: EXEC stays all-1s

  const int rowTile = waveId / nColTiles;
  const int colTile = waveId % nColTiles;

  const int r = (rowTile << 4) + (lane & 15);  // A row held by this lane
  const int c = (colTile << 4) + (lane & 15);  // B col / C col held by this lane
  const int khalf = (lane >> 4) << 1;          // lanes 16-31 hold K+2,K+3

  const float* __restrict__ arow = A  + (size_t)r * KDIM + khalf;
  const float* __restrict__ brow = BT + (size_t)c * KDIM + khalf;

  v8f acc = {};
  #pragma unroll
  for (int kk = 0; kk < KDIM; kk += 4) {
    const v2f a = *(const v2f*)(arow + kk);    // global_load_b64
    const v2f b = *(const v2f*)(brow + kk);    // global_load_b64
    // 8 args: (neg_a, A, neg_b, B, c_mod, C, reuse_a, reuse_b)
    acc = __builtin_amdgcn_wmma_f32_16x16x4_f32(false, a, false, b,
                                                (short)0, acc, false, false);
  }

  if (c < Ncol) {
    const int rbase = (rowTile << 4) + ((lane >> 4) << 3); // half-wave: M 0..7 / 8..15
    #pragma unroll
    for (int i = 0; i < 8; ++i)
      C[(size_t)(rbase + i) * Ncol + c] = acc[i];
  }
}

// ---------------------------------------------------------------------------
// Per-node attention logits: AS[n,h] = <H[n,h,:], a_src[h,:]>, same for AD.
// ---------------------------------------------------------------------------
__global__ void node_attn(const float* __restrict__ H,
                          const float* __restrict__ a_src,
                          const float* __restrict__ a_dst,
                          float* __restrict__ AS, float* __restrict__ AD,
                          int n, int heads, int chan)
{
  const long long idx = (long long)blockIdx.x * blockDim.x + threadIdx.x;
  const long long total = (long long)n * heads;
  if (idx >= total) return;
  const int node = (int)(idx / heads);
  const int h    = (int)(idx % heads);
  const float* __restrict__ hp = H + (size_t)node * heads * chan + (size_t)h * chan;
  const float* __restrict__ sv = a_src + (size_t)h * chan;
  const float* __restrict__ dv = a_dst + (size_t)h * chan;
  float s = 0.0f, d = 0.0f;
  for (int cidx = 0; cidx < chan; ++cidx) {
    const float v = hp[cidx];
    s += v * sv[cidx];
    d += v * dv[cidx];
  }
  AS[idx] = s;
  AD[idx] = d;
}

// ---------------------------------------------------------------------------
// Edge pass 1: EXs[e,h] = exp(leakyrelu(AS[src,h] + AD[dst,h])),
//              DEN[dst,h] += EXs[e,h].   Edges e>=E are self-loops (e-E,e-E).
// ---------------------------------------------------------------------------
__global__ void edge_num(const int* __restrict__ ei, int E, int n, int heads,
                         const float* __restrict__ AS, const float* __restrict__ AD,
                         float* __restrict__ EXs, float* __restrict__ DEN)
{
  const long long idx = (long long)blockIdx.x * blockDim.x + threadIdx.x;
  const long long tot = (long long)(E + n) * heads;
  if (idx >= tot) return;
  const int e = (int)(idx / heads);
  const int h = (int)(idx % heads);
  int src, dst;
  if (e < E) { src = ei[e]; dst = ei[E + e]; } else { src = dst = e - E; }
  float v = AS[(size_t)src * heads + h] + AD[(size_t)dst * heads + h];
  v = (v > 0.0f) ? v : NEG_SLOPE * v;
  const float ex = __expf(v);
  EXs[idx] = ex;
  atomAddF32(&DEN[(size_t)dst * heads + h], ex);
}

// ---------------------------------------------------------------------------
// Edge pass 2: OUT[dst, f] += H[src, f] * EXs[e,h]/DEN[dst,h], float4/thread.
// ---------------------------------------------------------------------------
__global__ void edge_agg(const int* __restrict__ ei, int E, int n, int heads, int chan,
                         const float* __restrict__ H, const float* __restrict__ EXs,
                         const float* __restrict__ DEN, float* __restrict__ OUT)
{
  const int F = heads * chan;
  const int Q = F >> 2;                         // float4 groups per edge (chan % 4 == 0)
  const long long idx = (long long)blockIdx.x * blockDim.x + threadIdx.x;
  const long long tot = (long long)(E + n) * Q;
  if (idx >= tot) return;
  const int e = (int)(idx / Q);
  const int q = (int)(idx % Q);
  const int f = q << 2;
  const int h = f / chan;
  int src, dst;
  if (e < E) { src = ei[e]; dst = ei[E + e]; } else { src = dst = e - E; }
  const float alpha = EXs[(size_t)e * heads + h] / DEN[(size_t)dst * heads + h];
  const float4 hv = *(const float4*)(H + (size_t)src * F + f);
  float* op = OUT + (size_t)dst * F + f;
  atomAddF32(op + 0, hv.x * alpha);
  atomAddF32(op + 1, hv.y * alpha);
  atomAddF32(op + 2, hv.z * alpha);
  atomAddF32(op + 3, hv.w * alpha);
}

__global__ void bias_init(float* __restrict__ out, const float* __restrict__ b,
                          int n, int F)
{
  const long long idx = (long long)blockIdx.x * blockDim.x + threadIdx.x;
  const long long tot = (long long)n * F;
  if (idx >= tot) return;
  out[idx] = b[idx % F];
}

__global__ void relu_inplace(float* __restrict__ x, long long n)
{
  const long long idx = (long long)blockIdx.x * blockDim.x + threadIdx.x;
  if (idx >= n) return;
  x[idx] = fmaxf(x[idx], 0.0f);
}

// ---------------------------------------------------------------------------
extern "C" void kernel_launch(void* const* d_in, const int* in_sizes, int n_in,
                              void* d_out, int out_size, void* d_ws, size_t ws_size,
                              hipStream_t stream)
{
  const float* x    = (const float*)d_in[0];
  const int*   ei   = (const int*)d_in[1];
  const float* W1   = (const float*)d_in[2];
  const float* as1w = (const float*)d_in[3];
  const float* ad1w = (const float*)d_in[4];
  const float* b1   = (const float*)d_in[5];
  const float* W2   = (const float*)d_in[6];
  const float* as2w = (const float*)d_in[7];
  const float* ad2w = (const float*)d_in[8];
  const float* b2   = (const float*)d_in[9];
  float* out = (float*)d_out;

  const int n  = in_sizes[0] / F_IN;     // 50000
  const int E  = in_sizes[1] / 2;        // 800000
  const int EP = E + n;                  // edges incl. self-loops

  // Workspace layout (floats)
  float* ws = (float*)d_ws;
  size_t off = 0;
  float* H1   = ws + off; off += (size_t)n * F_IN;        // x @ W1
  float* AGG1 = ws + off; off += (size_t)n * F_IN;        // layer-1 aggregation / h
  float* H2   = ws + off; off += (size_t)n * CLS;         // h @ W2
  float* WT1  = ws + off; off += (size_t)F_IN * KDIM;     // W1^T           [128x128]
  float* WT2  = ws + off; off += (size_t)CLSP * KDIM;     // W2^T, padded   [48x128]
  float* AS1  = ws + off; off += (size_t)n * HEADS;
  float* AD1  = ws + off; off += (size_t)n * HEADS;
  float* DEN1 = ws + off; off += (size_t)n * HEADS;
  float* AS2  = ws + off; off += (size_t)n;
  float* AD2  = ws + off; off += (size_t)n;
  float* DEN2 = ws + off; off += (size_t)n;
  float* EX1  = ws + off; off += (size_t)EP * HEADS;
  float* EX2  = ws + off; off += (size_t)EP;

  const int BS = 256;
  auto blocks = [](long long tot, int bs) { return (unsigned)((tot + bs - 1) / bs); };

  // Zero softmax denominators; seed aggregation buffers with biases.
  hipMemsetAsync(DEN1, 0, (size_t)n * HEADS * sizeof(float), stream);
  hipMemsetAsync(DEN2, 0, (size_t)n * sizeof(float), stream);
  bias_init<<<blocks((long long)n * F_IN, BS), BS, 0, stream>>>(AGG1, b1, n, F_IN);
  bias_init<<<blocks((long long)n * CLS,  BS), BS, 0, stream>>>(out,  b2, n, CLS);

  // Pre-transpose (and pad) weights so GEMM B-fragments are contiguous b64s.
  transpose_pad<<<blocks((long long)KDIM * F_IN, BS), BS, 0, stream>>>(
      W1, WT1, KDIM, F_IN, F_IN);
  transpose_pad<<<blocks((long long)KDIM * CLSP, BS), BS, 0, stream>>>(
      W2, WT2, KDIM, CLS, CLSP);

  // ---- Layer 1 ----
  {
    const int nColTiles = F_IN / 16;                          // 8
    const int tiles = (n / 16) * nColTiles;                   // 25000 waves
    wmma_gemm_f32<<<blocks((long long)tiles * 32, 256), 256, 0, stream>>>(
        x, WT1, H1, n, F_IN, nColTiles);
  }
  node_attn<<<blocks((long long)n * HEADS, BS), BS, 0, stream>>>(
      H1, as1w, ad1w, AS1, AD1, n, HEADS, HID);
  edge_num<<<blocks((long long)EP * HEADS, BS), BS, 0, stream>>>(
      ei, E, n, HEADS, AS1, AD1, EX1, DEN1);
  edge_agg<<<blocks((long long)EP * (F_IN / 4), BS), BS, 0, stream>>>(
      ei, E, n, HEADS, HID, H1, EX1, DEN1, AGG1);
  relu_inplace<<<blocks((long long)n * F_IN, BS), BS, 0, stream>>>(
      AGG1, (long long)n * F_IN);

  // ---- Layer 2 ----
  {
    const int nColTiles = CLSP / 16;                          // 3 (stores guarded)
    const int tiles = (n / 16) * nColTiles;                   // 9375 waves
    wmma_gemm_f32<<<blocks((long long)tiles * 32, 256), 256, 0, stream>>>(
        AGG1, WT2, H2, n, CLS, nColTiles);
  }
  node_attn<<<blocks((long long)n, BS), BS, 0, stream>>>(
      H2, as2w, ad2w, AS2, AD2, n, 1, CLS);
  edge_num<<<blocks((long long)EP, BS), BS, 0, stream>>>(
      ei, E, n, 1, AS2, AD2, EX2, DEN2);
  edge_agg<<<blocks((long long)EP * (CLS / 4), BS), BS, 0, stream>>>(
      ei, E, n, 1, CLS, H2, EX2, DEN2, out);
}